// DeepGate3_20547123544544
// MI455X (gfx1250) — compile-verified
//
#include <hip/hip_runtime.h>
#include <math.h>

#define D 128
#define CH 32            // members per chunk (2 x 16-row WMMA tiles per pool)
#define ROWPAD 132       // 128 + 4 floats: kills LDS bank conflicts on strided A reads

typedef float v2f __attribute__((ext_vector_type(2)));
typedef float v8f __attribute__((ext_vector_type(8)));

// ---------------------------------------------------------------------------
// Kernel 1: segment_ids is sorted -> per-gate ranges via binary search.
// seg_off[g] = lower_bound(segment_ids, g); seg_off[G] = E.
// ---------------------------------------------------------------------------
__global__ void seg_offsets_kernel(const int* __restrict__ seg, int E, int G,
                                   int* __restrict__ off) {
    int g = blockIdx.x * blockDim.x + threadIdx.x;
    if (g > G) return;
    int lo = 0, hi = E;
    while (lo < hi) {
        int mid = (lo + hi) >> 1;
        if (seg[mid] < g) lo = mid + 1; else hi = mid;
    }
    off[g] = lo;
}

// ---------------------------------------------------------------------------
// Kernel 2: one workgroup per gate. 128 threads = 4 waves (wave32).
// Fused dual-pool gather + WMMA logits + online softmax + weighted sum.
// ---------------------------------------------------------------------------
__global__ void __launch_bounds__(128)
attn_segment_pool_kernel(const float* __restrict__ tf_hs,
                         const float* __restrict__ tf_hf,
                         const float* __restrict__ w_hs,
                         const float* __restrict__ w_hf,
                         const int*   __restrict__ member_idx,
                         const int*   __restrict__ seg_off,
                         float* __restrict__ out, int G) {
    __shared__ __align__(16) float xs[2][CH][ROWPAD]; // staged rows, both pools
    __shared__ float wl[2][D];                        // weight vectors
    __shared__ float logit_s[2][CH];
    __shared__ float p_s[2][CH];
    __shared__ float mrun[2], zrun[2], scale_s[2];
    __shared__ int   idx_s[CH];

    const int g   = blockIdx.x;
    const int tid = threadIdx.x;
    const int wv  = tid >> 5;      // wave id 0..3
    const int ln  = tid & 31;      // lane id

    const int s0  = seg_off[g];
    const int cnt = seg_off[g + 1] - s0;

    wl[0][tid] = w_hs[tid];
    wl[1][tid] = w_hf[tid];
    if (tid == 0) {
        mrun[0] = -INFINITY; mrun[1] = -INFINITY;
        zrun[0] = 0.0f;      zrun[1] = 0.0f;
    }
    __syncthreads();

    float acc0 = 0.0f, acc1 = 0.0f;   // thread tid owns output dim = tid

    const float4* hs4 = (const float4*)tf_hs;
    const float4* hf4 = (const float4*)tf_hf;

    for (int base = 0; base < cnt; base += CH) {
        const int cc = min(CH, cnt - base);

        if (tid < CH) idx_s[tid] = (tid < cc) ? member_idx[s0 + base + tid] : -1;
        __syncthreads();

        // ---- gather: one wave per row, float4 -> 512B coalesced per row ----
        for (int jj = 0; jj < CH / 4; ++jj) {
            const int j  = jj * 4 + wv;
            const int ix = idx_s[j];
            float4 a = make_float4(0.f, 0.f, 0.f, 0.f);
            float4 b = a;
            if (ix >= 0) {
                a = hs4[(size_t)ix * (D / 4) + ln];
                b = hf4[(size_t)ix * (D / 4) + ln];
            }
            *(float4*)&xs[0][j][ln * 4] = a;
            *(float4*)&xs[1][j][ln * 4] = b;
        }
        __syncthreads();

        // ---- logits: wave -> (pool, 16-row half); f32 WMMA, w broadcast ----
        {
            const int pool    = wv >> 1;          // waves 0,1 -> hs ; 2,3 -> hf
            const int rowbase = (wv & 1) * 16;
            const float* row  = &xs[pool][rowbase + (ln & 15)][0];
            const float* wp   = &wl[pool][0];
            const int koff    = (ln & 16) ? 2 : 0;  // A/B 16x4 f32 lane->K map
#if defined(__gfx1250__) && __has_builtin(__builtin_amdgcn_wmma_f32_16x16x4_f32)
            v8f c = {};
            for (int kb = 0; kb < D / 4; ++kb) {
                const int k = kb * 4 + koff;
                v2f a;  a.x  = row[k]; a.y  = row[k + 1];
                v2f bb; bb.x = wp[k];  bb.y = wp[k + 1];
                c = __builtin_amdgcn_wmma_f32_16x16x4_f32(
                        false, a, false, bb, (short)0, c, false, false);
            }
            // D tile: VGPR r, lane 0 -> M=r, lane 16 -> M=r+8 (cols replicated)
            if ((ln & 15) == 0) {
                const int mhalf = (ln >> 4) * 8;
                for (int r = 0; r < 8; ++r)
                    logit_s[pool][rowbase + mhalf + r] = c[r];
            }
#else
            for (int mm = 0; mm < 16; ++mm) {
                const float* rr = &xs[pool][rowbase + mm][0];
                float part = 0.0f;
                for (int k = ln; k < D; k += 32) part = fmaf(rr[k], wp[k], part);
                for (int o = 16; o >= 1; o >>= 1) part += __shfl_xor(part, o, 32);
                if (ln == 0) logit_s[pool][rowbase + mm] = part;
            }
            (void)row; (void)koff;
#endif
        }
        __syncthreads();

        // ---- online softmax combine: wave0 = hs, wave1 = hf ----
        if (wv < 2) {
            const int pl = wv;
            float l = (ln < cc) ? logit_s[pl][ln] : -INFINITY;
            float mx = l;
            for (int o = 16; o >= 1; o >>= 1) mx = fmaxf(mx, __shfl_xor(mx, o, 32));
            const float mold = mrun[pl];
            const float mnew = fmaxf(mold, mx);
            const float p = (ln < cc) ? __expf(l - mnew) : 0.0f;
            p_s[pl][ln] = p;
            float s = p;
            for (int o = 16; o >= 1; o >>= 1) s += __shfl_xor(s, o, 32);
            if (ln == 0) {
                const float sc = __expf(mold - mnew);   // 0 on first chunk
                scale_s[pl] = sc;
                zrun[pl] = zrun[pl] * sc + s;
                mrun[pl] = mnew;
            }
        }
        __syncthreads();

        // ---- weighted accumulation: thread tid = dim tid, both pools ----
        acc0 *= scale_s[0];
        acc1 *= scale_s[1];
        for (int j = 0; j < cc; ++j) {
            acc0 = fmaf(p_s[0][j], xs[0][j][tid], acc0);
            acc1 = fmaf(p_s[1][j], xs[1][j][tid], acc1);
        }
        __syncthreads();
    }

    const float z0 = fmaxf(zrun[0], 1e-9f);
    const float z1 = fmaxf(zrun[1], 1e-9f);
    out[(size_t)g * D + tid]                 = acc0 / z0;   // hop_hs
    out[(size_t)G * D + (size_t)g * D + tid] = acc1 / z1;   // hop_hf
}

// ---------------------------------------------------------------------------
extern "C" void kernel_launch(void* const* d_in, const int* in_sizes, int n_in,
                              void* d_out, int out_size, void* d_ws, size_t ws_size,
                              hipStream_t stream) {
    const float* tf_hs      = (const float*)d_in[0];
    const float* tf_hf      = (const float*)d_in[1];
    const float* w_hs       = (const float*)d_in[2];
    const float* w_hf       = (const float*)d_in[3];
    const int*   member_idx = (const int*)d_in[4];
    const int*   seg_ids    = (const int*)d_in[5];

    const int E = in_sizes[4];
    const int G = out_size / (2 * D);

    int* seg_off = (int*)d_ws;   // (G+1) ints of scratch

    seg_offsets_kernel<<<(G + 1 + 255) / 256, 256, 0, stream>>>(seg_ids, E, G, seg_off);
    attn_segment_pool_kernel<<<G, 128, 0, stream>>>(
        tf_hs, tf_hf, w_hs, w_hf, member_idx, seg_off, (float*)d_out, G);
}